// eca_layer_66211215835519
// MI455X (gfx1250) — compile-verified
//
#include <hip/hip_runtime.h>
#include <hip/hip_bf16.h>
#include <stdint.h>

// ECA layer: x(16,1024,64,64) f32, conv_w(1,1,3) f32
//   mean over HW -> dilated 3-tap conv (D=510,pad=2) -> sigmoid (16x8 gates)
//   out[b,c,h,w] = x[b,c,h,w] * s[b, c>>7]
//
// Memory-bound: 768 MB traffic @ 23.3 TB/s ~= 33 us floor. AI ~ 0.26 F/B ->
// WMMA is useless here; optimize the data path instead:
//  - pass 1 streams x via gfx1250 async-to-LDS b128 copies (ASYNCcnt)
//  - pass 3 uses non-temporal load/store so the 256MB output stream does not
//    evict x from the 192MB L2 (x is ~75% L2-resident from pass 1)

typedef float v4f __attribute__((ext_vector_type(4)));

#define CHANNEL   1024
#define BATCH     16
#define HW        4096            // 64*64
#define NBC       (BATCH * CHANNEL)   // 16384 (b,c) rows
#define NGATE     (BATCH * 8)         // 128 sigmoid gates

// ---------------------------------------------------------------------------
// Pass 1: per-(b,c) mean over 4096 floats. One 256-thread block per row.
// Each lane async-copies 4 x 16B of its row into LDS (the offset: immediate is
// added to BOTH the LDS and global address, so one base pair + 3 offsets
// mirrors the row layout exactly), waits on ASYNCcnt, sums its own staged
// float4s, then a standard LDS tree reduction finishes the row.
// ---------------------------------------------------------------------------
__global__ __launch_bounds__(256) void eca_reduce_mean_kernel(
    const float* __restrict__ x, float* __restrict__ mean) {
  __shared__ v4f   tile4[HW / 4];   // 16 KB staged row
  __shared__ float red[256];

  const int tid = threadIdx.x;
  const int bc  = blockIdx.x;
  const float* src = x + (size_t)bc * HW;

#if defined(__HIP_DEVICE_COMPILE__)
  {
    // low 32 bits of the generic (shared-aperture) address == LDS byte offset
    unsigned laddr = (unsigned)(uintptr_t)tile4 + (unsigned)tid * 16u;
    uint64_t gaddr = (uint64_t)(uintptr_t)src + (uint64_t)tid * 16u;
    asm volatile(
        "global_load_async_to_lds_b128 %0, %1, off\n\t"
        "global_load_async_to_lds_b128 %0, %1, off offset:4096\n\t"
        "global_load_async_to_lds_b128 %0, %1, off offset:8192\n\t"
        "global_load_async_to_lds_b128 %0, %1, off offset:12288\n\t"
        "s_wait_asynccnt 0x0"
        :: "v"(laddr), "v"(gaddr)
        : "memory");
  }
#endif

  // Each lane reads back exactly the 4 float4s it staged itself, so the
  // per-wave s_wait_asynccnt above is the only ordering needed before these.
  v4f a0 = tile4[tid];
  v4f a1 = tile4[tid + 256];
  v4f a2 = tile4[tid + 512];
  v4f a3 = tile4[tid + 768];
  v4f s4 = (a0 + a1) + (a2 + a3);
  red[tid] = s4.x + s4.y + s4.z + s4.w;
  __syncthreads();

  for (int s = 128; s > 0; s >>= 1) {
    if (tid < s) red[tid] += red[tid + s];
    __syncthreads();
  }
  if (tid == 0) mean[bc] = red[0] * (1.0f / (float)HW);
}

// ---------------------------------------------------------------------------
// Pass 2: dilated conv (D=510, pad=2) + sigmoid. Output length is 8 per batch:
//   s[b,p] = sigmoid( w0*m[p-2] (p>=2) + w1*m[p+508] + w2*m[p+1018] (p<=5) )
// 128 values total -> one tiny block.
// ---------------------------------------------------------------------------
__global__ __launch_bounds__(128) void eca_gate_kernel(
    const float* __restrict__ mean, const float* __restrict__ w,
    float* __restrict__ gate) {
  const int t = threadIdx.x;          // 0..127
  const int b = t >> 3;
  const int p = t & 7;
  const float* m = mean + b * CHANNEL;

  float acc = w[1] * m[p + 508];
  if (p >= 2) acc += w[0] * m[p - 2];
  if (p <= 5) acc += w[2] * m[p + 1018];
  gate[t] = 1.0f / (1.0f + expf(-acc));
}

// ---------------------------------------------------------------------------
// Pass 3: out = x * gate[b, c>>7]. Flat float4 index i4 -> element e = i4*4;
// gate index = b*8 + (c>>7) = e>>19 = i4>>17.
// Non-temporal on both streams: x is last-use (keep L2 hits, don't refill),
// out is write-once (don't let 256MB of stores evict x mid-pass).
// ---------------------------------------------------------------------------
__global__ __launch_bounds__(256) void eca_scale_kernel(
    const v4f* __restrict__ x4, const float* __restrict__ gate,
    v4f* __restrict__ out4) {
  const size_t i = (size_t)blockIdx.x * 256 + threadIdx.x;
  const float g = gate[i >> 17];
  v4f v = __builtin_nontemporal_load(&x4[i]);
  __builtin_nontemporal_store(v * g, &out4[i]);
}

// ---------------------------------------------------------------------------
extern "C" void kernel_launch(void* const* d_in, const int* in_sizes, int n_in,
                              void* d_out, int out_size, void* d_ws, size_t ws_size,
                              hipStream_t stream) {
  const float* x = (const float*)d_in[0];   // 16*1024*64*64
  const float* w = (const float*)d_in[1];   // 3
  float* out  = (float*)d_out;              // 16*1024*64*64
  float* mean = (float*)d_ws;               // NBC floats
  float* gate = mean + NBC;                 // NGATE floats

  eca_reduce_mean_kernel<<<NBC, 256, 0, stream>>>(x, mean);
  eca_gate_kernel<<<1, 128, 0, stream>>>(mean, w, gate);

  const int n4 = out_size / 4;              // 16,777,216 float4s
  eca_scale_kernel<<<n4 / 256, 256, 0, stream>>>((const v4f*)x, gate, (v4f*)out);
}